// SuperResolution_23115513987123
// MI455X (gfx1250) — compile-verified
//
#include <hip/hip_runtime.h>

// Super-resolution degradation: per-sample block-mean (k = 2^t, k in {1,2,4,8})
// broadcast over each k x k block. Pure bandwidth problem: 402 MB traffic ->
// ~17 us floor at 23.3 TB/s. Data movement via CDNA5 async global->LDS DMA
// (ASYNCcnt) instead of VGPR round-trips; wave32, 8-wave blocks.

#define IMG_W       256
#define STRIP_ROWS  8
#define STRIP_ELEMS (STRIP_ROWS * IMG_W)   // 2048 floats = 8 KB per strip
#define THREADS     256                    // 8 wave32 waves

#ifndef __has_builtin
#define __has_builtin(x) 0
#endif

__device__ __forceinline__ void async_copy_b128(const void* g, void* l) {
#if __has_builtin(__builtin_amdgcn_global_load_async_to_lds_b128)
  // Toolchain signature (from round-1 diagnostic):
  //   void (int4 __device__* gsrc, int4 __shared__* ldst, imm int off, imm int cpol)
  typedef int v4i __attribute__((vector_size(16)));
  typedef __attribute__((address_space(1))) v4i* gp_t;   // global (AS1, 64-bit)
  typedef __attribute__((address_space(3))) v4i* lp_t;   // LDS    (AS3, 32-bit)
  gp_t gp = (gp_t)(unsigned long long)g;
  lp_t lp = (lp_t)(unsigned int)(unsigned long long)l;   // low 32b = LDS offset
  __builtin_amdgcn_global_load_async_to_lds_b128(gp, lp, 0, 0);
#else
  typedef __attribute__((address_space(3))) char LCH;
  unsigned lofs = (unsigned)(unsigned long long)(LCH*)l;  // LDS byte offset
  asm volatile("global_load_async_to_lds_b128 %0, %1, off"
               :: "v"(lofs), "v"(g) : "memory");
#endif
}

__device__ __forceinline__ void wait_asynccnt0() {
#if __has_builtin(__builtin_amdgcn_s_wait_asynccnt)
  __builtin_amdgcn_s_wait_asynccnt(0);
#else
  asm volatile("s_wait_asynccnt 0x0" ::: "memory");
#endif
}

__global__ __launch_bounds__(THREADS)
void SuperRes_degrade_kernel(const float* __restrict__ x0,
                             const int*   __restrict__ tlev,
                             float*       __restrict__ out) {
  __shared__ float tile[STRIP_ELEMS];   // 8 KB strip
  __shared__ float colsum[4 * IMG_W];   // max 8/k = 4 block-rows (k=2)
  __shared__ float bmean[512];          // max 4 * 128 means (k=2)

  const int tid = threadIdx.x;
  const int sid = blockIdx.x;           // strip id; 96 strips per batch sample
  const int b   = sid / 96;             // 3 channels * 32 strips
  const size_t base = (size_t)sid * STRIP_ELEMS;

  int tv = tlev[b];
  tv = tv < 0 ? 0 : (tv > 3 ? 3 : tv);  // uniform across the whole block

  // ---- async-stage strip (8192 B = 512 x 16 B) into LDS: 2 chunks/thread ----
  {
    const char* g = (const char*)(x0 + base);
    char*       l = (char*)tile;
    async_copy_b128(g + tid * 16,        l + tid * 16);
    async_copy_b128(g + 4096 + tid * 16, l + 4096 + tid * 16);
    wait_asynccnt0();                   // my async loads done
  }
  __syncthreads();                      // everyone's async loads visible

  if (tv > 0) {
    const int k   = 1 << tv;
    const int nbr = STRIP_ROWS >> tv;   // block-rows in strip
    const int nbc = IMG_W >> tv;        // block-cols

    // phase 1: vertical k-row sums, thread <-> column (bank-conflict free)
    for (int br = 0; br < nbr; ++br) {
      const float* row = &tile[br * k * IMG_W + tid];
      float s = 0.0f;
      for (int r = 0; r < k; ++r) s += row[r * IMG_W];
      colsum[br * IMG_W + tid] = s;
    }
    __syncthreads();

    // phase 2: horizontal k-sums -> block means
    const int   nblocks = nbr * nbc;    // 512 / 128 / 32 for k=2/4/8
    const float inv     = 1.0f / (float)(k * k);
    for (int i = tid; i < nblocks; i += THREADS) {
      const int br = i / nbc;
      const int bc = i - br * nbc;
      const float* cs = &colsum[br * IMG_W + bc * k];
      float s = 0.0f;
      for (int j = 0; j < k; ++j) s += cs[j];
      bmean[br * nbc + bc] = s * inv;
    }
    __syncthreads();
  }

  // ---- write phase: each thread stores 2 rows x 4 cols as float4 (b128) ----
  const int c4 = (tid & 63) * 4;        // column group, 16 B aligned
  const int r0 = (tid >> 6) * 2;        // row pair
  float* dst = out + base;
#pragma unroll
  for (int dr = 0; dr < 2; ++dr) {
    const int r = r0 + dr;
    float4 v;
    if (tv == 0) {
      v = *(const float4*)&tile[r * IMG_W + c4];        // identity level
    } else {
      const int nbc = IMG_W >> tv;
      const float* m = &bmean[(r >> tv) * nbc];         // broadcast reads
      v.x = m[(c4 + 0) >> tv];
      v.y = m[(c4 + 1) >> tv];
      v.z = m[(c4 + 2) >> tv];
      v.w = m[(c4 + 3) >> tv];
    }
    *(float4*)&dst[r * IMG_W + c4] = v;                 // coalesced b128 store
  }
}

extern "C" void kernel_launch(void* const* d_in, const int* in_sizes, int n_in,
                              void* d_out, int out_size, void* d_ws, size_t ws_size,
                              hipStream_t stream) {
  (void)in_sizes; (void)n_in; (void)d_ws; (void)ws_size;
  const float* x0 = (const float*)d_in[0];
  const int*   t  = (const int*)d_in[1];   // per harness: integer -> const int*
  float*       o  = (float*)d_out;

  const int nstrips = out_size / STRIP_ELEMS;  // 256*3*256*256 / 2048 = 24576
  SuperRes_degrade_kernel<<<nstrips, THREADS, 0, stream>>>(x0, t, o);
}